// MDNLoss_83648783057105
// MI455X (gfx1250) — compile-verified
//
#include <hip/hip_runtime.h>
#include <math.h>

// Problem constants (from reference)
#define B_      8
#define TXT_T   200
#define TXT_PAD 208        // pad t to multiple of 16 for WMMA tiling
#define MEL_T   800
#define N_MEL   80
#define KDIM    160        // 2*N_MEL packed contraction dim
#define PAD_MAG 1.0e12f

typedef __attribute__((ext_vector_type(2))) float v2f;
typedef __attribute__((ext_vector_type(8))) float v8f;

// Workspace layout (floats):
//   Aprep [B][TXT_PAD][KDIM]  : A'[t,k] = {w, -2*mu*w}
//   Cprep [B][TXT_PAD]        : C[t]    = sum_n (mu^2*w + lv)
//   partial [B]               : per-batch alpha_last / mel_len
#define A_OFF 0
#define C_OFF (B_ * TXT_PAD * KDIM)
#define P_OFF (C_OFF + B_ * TXT_PAD)

// ---------------------------------------------------------------- prep ------
__global__ void mdn_prep_kernel(const float* __restrict__ mu_logvar,
                                float* __restrict__ ws) {
  int idx = blockIdx.x * blockDim.x + threadIdx.x;
  if (idx >= B_ * TXT_PAD) return;
  int b = idx / TXT_PAD;
  int t = idx % TXT_PAD;
  float* Arow = ws + A_OFF + (size_t)idx * KDIM;
  float c = 0.0f;
  if (t < TXT_T) {
    const float* ml = mu_logvar + (size_t)(b * TXT_T + t) * (2 * N_MEL);
    for (int n = 0; n < N_MEL; ++n) {
      float mu = ml[n];
      float lv = ml[N_MEL + n];
      float w  = expf(-lv);
      Arow[n]          = w;
      Arow[N_MEL + n]  = -2.0f * mu * w;
      c += mu * mu * w + lv;
    }
  } else {
    for (int n = 0; n < KDIM; ++n) Arow[n] = 0.0f;   // zero pad rows
  }
  ws[C_OFF + idx] = c;
}

// ---------------------------------------------------------------- gemm ------
// One wave32 per 16(t) x 16(m) output tile; K-loop of 40 x V_WMMA_F32_16X16X4_F32.
// A operand (16x4 f32): lanes 0-15 -> rows t0..t0+15 with K = k,k+1 in the two
// VGPRs; lanes 16-31 -> same rows with K = k+2,k+3. B operand mirrored:
// VGPR0/1 hold K = k+2*klane {+1}, N = m0 + (lane&15).
__global__ void __launch_bounds__(256)
mdn_gemm_kernel(const float* __restrict__ ws,
                const float* __restrict__ mel,   // [B][N_MEL][MEL_T]
                float* __restrict__ lp) {        // [B][TXT_T][MEL_T]
  const int TT = TXT_PAD / 16;   // 13 t-tiles
  const int MT = MEL_T / 16;     // 50 m-tiles
  int wid = blockIdx.x * (blockDim.x >> 5) + (threadIdx.x >> 5);
  if (wid >= B_ * TT * MT) return;           // uniform per wave (grid is exact)
  int lane = threadIdx.x & 31;
  int l15  = lane & 15;
  int kl   = lane >> 4;

  int b  = wid / (TT * MT);
  int r  = wid % (TT * MT);
  int t0 = (r / MT) * 16;
  int m0 = (r % MT) * 16;

  const float* Arow = ws + A_OFF + (size_t)(b * TXT_PAD + t0 + l15) * KDIM;
  const float* melb = mel + (size_t)b * N_MEL * MEL_T + m0 + l15;

  v8f acc = {0.f, 0.f, 0.f, 0.f, 0.f, 0.f, 0.f, 0.f};

  for (int k = 0; k < KDIM; k += 4) {
    v2f a = *(const v2f*)(Arow + k + 2 * kl);

    int K0 = k + 2 * kl;                      // K handled by this half-wave
    int n  = (k < N_MEL) ? K0 : (K0 - N_MEL); // k never straddles the 80 split
    float x0 = melb[(size_t)n * MEL_T];
    float x1 = melb[(size_t)(n + 1) * MEL_T];
    if (k < N_MEL) { x0 *= x0; x1 *= x1; }    // first 80 K's carry x^2
    v2f bv; bv.x = x0; bv.y = x1;

    acc = __builtin_amdgcn_wmma_f32_16x16x4_f32(
        /*neg_a=*/false, a, /*neg_b=*/false, bv,
        /*c_mod=*/(short)0, acc, /*reuse_a=*/false, /*reuse_b=*/false);
  }

  // D layout: VGPR j -> row t0+j (lanes 0-15) / t0+8+j (lanes 16-31); col = m0+l15.
  int m = m0 + l15;
  const float* Cb = ws + C_OFF + b * TXT_PAD;
  const float scale = -0.5f / (float)N_MEL;
#pragma unroll
  for (int j = 0; j < 8; ++j) {
    int t = t0 + j + 8 * kl;
    if (t < TXT_T) {
      lp[(size_t)(b * TXT_T + t) * MEL_T + m] = scale * (acc[j] + Cb[t]);
    }
  }
}

// ---------------------------------------------------------------- scan ------
__device__ __forceinline__ float logaddexp_(float u, float v) {
  float mx = fmaxf(u, v);
  float mn = fminf(u, v);
  return mx + log1pf(expf(mn - mx));
}

__global__ void mdn_scan_kernel(const float* __restrict__ lp,
                                const int* __restrict__ text_len,
                                const int* __restrict__ mel_len,
                                float* __restrict__ ws) {
  __shared__ float sa[TXT_T + 1];
  int b = blockIdx.x;
  int t = threadIdx.x;
  for (int i = t; i <= TXT_T; i += blockDim.x) sa[i] = -PAD_MAG;
  __syncthreads();
  const float* lpb = lp + (size_t)b * TXT_T * MEL_T;
  if (t == 0) sa[1] = lpb[0];                 // log_alpha[b,0,0]
  __syncthreads();

  int M = mel_len[b];
  for (int m = 1; m < M; ++m) {
    float u = 0.f, v = 0.f;
    if (t < TXT_T) { u = sa[t + 1]; v = sa[t]; }
    __syncthreads();
    if (t < TXT_T) {
      sa[t + 1] = logaddexp_(u + 1e-7f, v + 1e-7f) + lpb[(size_t)t * MEL_T + m];
    }
    __syncthreads();
  }
  if (t == 0) {
    int tl = text_len[b];                     // 1-based length -> sa[tl]
    ws[P_OFF + b] = sa[tl] / (float)M;
  }
}

__global__ void mdn_reduce_kernel(const float* __restrict__ ws,
                                  float* __restrict__ out) {
  float s = 0.f;
  for (int b = 0; b < B_; ++b) s += ws[P_OFF + b];
  out[0] = -s / (float)B_;
}

// -------------------------------------------------------------- launch ------
extern "C" void kernel_launch(void* const* d_in, const int* in_sizes, int n_in,
                              void* d_out, int out_size, void* d_ws, size_t ws_size,
                              hipStream_t stream) {
  const float* mu_logvar = (const float*)d_in[0];
  const float* melspec   = (const float*)d_in[1];
  const int*   text_len  = (const int*)d_in[2];
  const int*   mel_len   = (const int*)d_in[3];
  float* out = (float*)d_out;
  float* ws  = (float*)d_ws;
  float* lp  = out + 1;   // [B, TXT_T, MEL_T] flattened after the scalar loss

  mdn_prep_kernel<<<(B_ * TXT_PAD + 255) / 256, 256, 0, stream>>>(mu_logvar, ws);

  const int total_waves = B_ * (TXT_PAD / 16) * (MEL_T / 16);  // 5200
  mdn_gemm_kernel<<<total_waves / 8, 256, 0, stream>>>(ws, melspec, lp);

  mdn_scan_kernel<<<B_, 256, 0, stream>>>(lp, text_len, mel_len, ws);
  mdn_reduce_kernel<<<1, 1, 0, stream>>>(ws, out);
}